// CompetitiveInhibitorLayer_21474836480242
// MI455X (gfx1250) — compile-verified
//
#include <hip/hip_runtime.h>

#define B_ 4
#define S_ 4096
#define H_ 2048
#define NROWS (B_ * S_)
#define DECAY_F 0.95f
#define SCAN_THREADS 128
#define CHUNK 32
#define THREADS2 256

typedef float v4f __attribute__((ext_vector_type(4)));
typedef int   vi4 __attribute__((vector_size(16)));   // matches builtin param type

__device__ __forceinline__ float hit_of(int v) {
  return (v == 5 || v == 13 || v == 42 || v == 99) ? 1.0f : 0.0f;
}

// ---------------------------------------------------------------------------
// Kernel A (listed first so the disasm snippet shows it):
// one block per (b,s) row of H=2048 floats.
//   - async global->LDS copy of the row (global_load_async_to_lds_b128, ASYNCcnt)
//   - wave32 shuffle + LDS reduction for sum of squares
//   - payload anomaly flag, combine with trigger mask, scale, NT b128 store
// ---------------------------------------------------------------------------
__global__ __launch_bounds__(THREADS2)
void inhibit_kernel(const float* __restrict__ hs,
                    const float* __restrict__ trig,
                    const float* __restrict__ w,
                    float* __restrict__ out) {
  __shared__ __align__(16) float row[H_];
  __shared__ float red[THREADS2 / 32];
  __shared__ float ss_sh;

  const int r = blockIdx.x;          // row in [0, B_*S_)
  const int t = threadIdx.x;
  const size_t rb = (size_t)r * H_;
  const float* src = hs + rb;

  // --- async copy: each lane issues two b128 loads into LDS ----------------
#if __has_builtin(__builtin_amdgcn_global_load_async_to_lds_b128)
  __builtin_amdgcn_global_load_async_to_lds_b128(
      (vi4*)(src + t * 4), (vi4*)(row + t * 4), 0, 0);
  __builtin_amdgcn_global_load_async_to_lds_b128(
      (vi4*)(src + 1024 + t * 4), (vi4*)(row + 1024 + t * 4), 0, 0);
#else
  {
    unsigned lds0 = (unsigned)(size_t)(row + t * 4);
    unsigned lds1 = (unsigned)(size_t)(row + 1024 + t * 4);
    unsigned long long g0 = (unsigned long long)(size_t)(src + t * 4);
    unsigned long long g1 = (unsigned long long)(size_t)(src + 1024 + t * 4);
    asm volatile("global_load_async_to_lds_b128 %0, %1, off"
                 :: "v"(lds0), "v"(g0) : "memory");
    asm volatile("global_load_async_to_lds_b128 %0, %1, off"
                 :: "v"(lds1), "v"(g1) : "memory");
  }
#endif
#if __has_builtin(__builtin_amdgcn_s_wait_asynccnt)
  __builtin_amdgcn_s_wait_asynccnt(0);
#else
  asm volatile("s_wait_asynccnt 0" ::: "memory");
#endif
  asm volatile("" ::: "memory");   // compiler barrier: LDS now holds the row

  v4f x0 = *(const v4f*)(row + t * 4);
  v4f x1 = *(const v4f*)(row + 1024 + t * 4);

  float p = x0[0] * x0[0] + x0[1] * x0[1] + x0[2] * x0[2] + x0[3] * x0[3]
          + x1[0] * x1[0] + x1[1] * x1[1] + x1[2] * x1[2] + x1[3] * x1[3];

  // wave32 reduction
#pragma unroll
  for (int o = 16; o > 0; o >>= 1) p += __shfl_xor(p, o, 32);

  const int wave = t >> 5;
  if ((t & 31) == 0) red[wave] = p;
  __syncthreads();
  if (t == 0) {
    float s = 0.0f;
#pragma unroll
    for (int i = 0; i < THREADS2 / 32; ++i) s += red[i];
    ss_sh = s;
  }
  __syncthreads();

  const float ss = ss_sh;
  const float denom = fmaxf(sqrtf(ss), 1e-12f);
  const float diag = ss / (denom * denom);
  const float anomaly = 1.0f - fminf(fmaxf(diag, 0.0f), 1.0f);
  const float payload = (anomaly > 0.5f) ? 1.0f : 0.0f;

  const float combined = fmaxf(trig[r], payload);
  const float mask = 1.0f - combined * w[0];

  if (t == 0) out[(size_t)NROWS * H_ + r] = mask;  // inhibition_mask tail

  v4f y0 = x0 * mask;
  v4f y1 = x1 * mask;
  // streaming output: non-temporal b128 stores (output is never re-read)
  __builtin_nontemporal_store(y0, (v4f*)(out + rb + t * 4));
  __builtin_nontemporal_store(y1, (v4f*)(out + rb + 1024 + t * 4));
}

// ---------------------------------------------------------------------------
// Kernel B: per-batch decay-max scan along S.
//   m[0] = hit[0]; m[t] = max(hit[t], 0.95*m[t-1])
// Parallelized via the associative transform (v,f): s -> max(v, s*f).
// grid = B_, block = SCAN_THREADS, each thread owns a CHUNK-long segment.
// ---------------------------------------------------------------------------
__global__ __launch_bounds__(SCAN_THREADS)
void trigger_scan_kernel(const int* __restrict__ ids, float* __restrict__ trig) {
  __shared__ float sv[SCAN_THREADS];
  __shared__ float sf[SCAN_THREADS];
  const int b = blockIdx.x;
  const int t = threadIdx.x;
  const int base = b * S_ + t * CHUNK;

  // Pass 1: local scan with zero carry-in; f = 0.95^CHUNK via iterated mul.
  float m = 0.0f, f = 1.0f;
#pragma unroll 8
  for (int k = 0; k < CHUNK; ++k) {
    float h = hit_of(ids[base + k]);
    m = fmaxf(h, m * DECAY_F);
    f *= DECAY_F;
  }
  sv[t] = m;
  sf[t] = f;
  __syncthreads();

  // Hillis-Steele inclusive scan over the chunk transforms.
  for (int off = 1; off < SCAN_THREADS; off <<= 1) {
    float pv = 0.0f, pf = 1.0f;
    if (t >= off) { pv = sv[t - off]; pf = sf[t - off]; }
    __syncthreads();
    if (t >= off) {
      sv[t] = fmaxf(sv[t], pv * sf[t]);  // compose: older-first
      sf[t] = pf * sf[t];
    }
    __syncthreads();
  }

  // Carry into this chunk = inclusive result of chunks [0, t).
  float carry = (t == 0) ? 0.0f : sv[t - 1];

  // Pass 2: redo local scan with real carry, emit trigger mask.
  m = carry;
#pragma unroll 8
  for (int k = 0; k < CHUNK; ++k) {
    float h = hit_of(ids[base + k]);
    m = fmaxf(h, m * DECAY_F);
    trig[base + k] = m;
  }
}

// ---------------------------------------------------------------------------
extern "C" void kernel_launch(void* const* d_in, const int* in_sizes, int n_in,
                              void* d_out, int out_size, void* d_ws, size_t ws_size,
                              hipStream_t stream) {
  const float* hs  = (const float*)d_in[0];   // hidden_states (4,4096,2048) f32
  const int*   ids = (const int*)d_in[1];     // input_ids (4,4096) int32 (JAX x64 off)
  const float* w   = (const float*)d_in[2];   // inhibitor_weight [1] f32
  float* out  = (float*)d_out;                // [B*S*H inhibited | B*S mask]
  float* trig = (float*)d_ws;                 // B*S floats scratch (64 KB)

  trigger_scan_kernel<<<B_, SCAN_THREADS, 0, stream>>>(ids, trig);
  inhibit_kernel<<<NROWS, THREADS2, 0, stream>>>(hs, trig, w, out);
}